// pose_regression_CMA_52570399703187
// MI455X (gfx1250) — compile-verified
//
#include <hip/hip_runtime.h>
#include <math.h>

typedef __attribute__((ext_vector_type(2))) float v2f;
typedef __attribute__((ext_vector_type(8))) float v8f;

__device__ __forceinline__ float mish_f(float v) {
  // mish(v) = v * tanh(softplus(v)); softplus(v)≈v for large v (avoid exp overflow)
  float sp = (v > 20.0f) ? v : log1pf(__expf(v));
  return v * tanhf(sp);
}

__global__ void mish_kernel(const float* __restrict__ in, float* __restrict__ out, int n) {
  int i = blockIdx.x * blockDim.x + threadIdx.x;
  if (i < n) out[i] = mish_f(in[i]);
}

// out[b, 0:2N] = mish( concat(xs[b,:], ys[b,:]) + (dprev ? dprev[b,:] : 0) )
__global__ void cat_mish_kernel(const float* __restrict__ xs, const float* __restrict__ ys,
                                const float* __restrict__ dprev, float* __restrict__ out,
                                int N, int B) {
  int twoN = 2 * N;
  int idx = blockIdx.x * blockDim.x + threadIdx.x;
  if (idx >= B * twoN) return;
  int b = idx / twoN;
  int c = idx - b * twoN;
  float v = (c < N) ? xs[b * N + c] : ys[b * N + (c - N)];
  if (dprev) v += dprev[idx];
  out[idx] = mish_f(v);
}

// C[m,n] = sum_k A[m,k] * W[n,k] (+ bias[n])  via V_WMMA_F32_16X16X4_F32.
// One wave computes a 32x32 output tile (2x2 of 16x16): per K-step of 4 we do
// 4 b64 loads and 4 WMMAs into 4 independent accumulators (1:1 load:wmma, no
// RAW chain within a step).
// A-layout (ISA 7.12.2, 32-bit A 16x4): lane&15 = M, (lane>=16 ? K+2 : K) in v0/v1.
// B-layout mirrored (lane&15 = N). C/D: VGPR v -> M = v (+8 for lanes>=16), N = lane&15.
__global__ void gemm_wmma_f32(const float* __restrict__ A, const float* __restrict__ W,
                              const float* __restrict__ bias, float* __restrict__ C,
                              int M, int N, int K) {
  int wave = threadIdx.x >> 5;
  int lane = threadIdx.x & 31;
  int tiles_n = N >> 5;                        // 32-wide n tiles
  int tile = blockIdx.x * 4 + wave;
  if (tile >= (M >> 5) * tiles_n) return;      // wave-uniform guard: EXEC stays all-1s
  int tm = tile / tiles_n;
  int m0 = tm << 5;
  int n0 = (tile - tm * tiles_n) << 5;
  int r  = lane & 15;
  int kh = (lane >> 4) << 1;                   // 0 for lanes 0-15, 2 for lanes 16-31
  const float* Ar0 = A + (size_t)(m0 + r) * K + kh;
  const float* Ar1 = Ar0 + (size_t)16 * K;
  const float* Wr0 = W + (size_t)(n0 + r) * K + kh;
  const float* Wr1 = Wr0 + (size_t)16 * K;
  v8f acc00 = {};
  v8f acc01 = {};
  v8f acc10 = {};
  v8f acc11 = {};
  for (int k = 0; k < K; k += 4) {             // K is always a multiple of 4 here
    v2f a0, a1, b0, b1;
    a0.x = Ar0[k]; a0.y = Ar0[k + 1];
    a1.x = Ar1[k]; a1.y = Ar1[k + 1];
    b0.x = Wr0[k]; b0.y = Wr0[k + 1];
    b1.x = Wr1[k]; b1.y = Wr1[k + 1];
    acc00 = __builtin_amdgcn_wmma_f32_16x16x4_f32(false, a0, false, b0, (short)0, acc00, false, false);
    acc01 = __builtin_amdgcn_wmma_f32_16x16x4_f32(false, a0, false, b1, (short)0, acc01, false, false);
    acc10 = __builtin_amdgcn_wmma_f32_16x16x4_f32(false, a1, false, b0, (short)0, acc10, false, false);
    acc11 = __builtin_amdgcn_wmma_f32_16x16x4_f32(false, a1, false, b1, (short)0, acc11, false, false);
  }
  int n_lo = n0 + r;
  int n_hi = n_lo + 16;
  float bv0 = bias ? bias[n_lo] : 0.0f;
  float bv1 = bias ? bias[n_hi] : 0.0f;
  int mb = m0 + ((lane >> 4) << 3);
#pragma unroll
  for (int v = 0; v < 8; ++v) {
    C[(size_t)(mb + v) * N + n_lo]      = acc00[v] + bv0;
    C[(size_t)(mb + v) * N + n_hi]      = acc01[v] + bv1;
    C[(size_t)(mb + 16 + v) * N + n_lo] = acc10[v] + bv0;
    C[(size_t)(mb + 16 + v) * N + n_hi] = acc11[v] + bv1;
  }
}

// out[b,i] = self[b,i] + beta * sum_j softmax_j(self_i * other_j) * self_j
// beta == 0 (the provided inputs) short-circuits to a copy.
__global__ void attn_kernel(const float* __restrict__ self, const float* __restrict__ other,
                            const float* __restrict__ beta_p, float* __restrict__ out, int N) {
  __shared__ float sh_s[512];
  __shared__ float sh_o[512];
  __shared__ float smx[256];
  __shared__ float smn[256];
  int b = blockIdx.x;
  int t = threadIdx.x;
  const float* s = self  + (size_t)b * N;
  const float* o = other + (size_t)b * N;
  float beta = beta_p[0];
  if (beta == 0.0f) {
    for (int i = t; i < N; i += blockDim.x) out[(size_t)b * N + i] = s[i];
    return;
  }
  for (int i = t; i < N; i += blockDim.x) { sh_s[i] = s[i]; sh_o[i] = o[i]; }
  __syncthreads();
  float mx = -3.4e38f, mn = 3.4e38f;
  for (int i = t; i < N; i += blockDim.x) { mx = fmaxf(mx, sh_o[i]); mn = fminf(mn, sh_o[i]); }
  smx[t] = mx; smn[t] = mn;
  __syncthreads();
  for (int off = 128; off > 0; off >>= 1) {
    if (t < off) { smx[t] = fmaxf(smx[t], smx[t + off]); smn[t] = fminf(smn[t], smn[t + off]); }
    __syncthreads();
  }
  float omax = smx[0], omin = smn[0];
  for (int i = t; i < N; i += blockDim.x) {
    float si = sh_s[i];
    float m = (si >= 0.0f) ? si * omax : si * omin;   // max of rank-1 logits
    float den = 0.0f, num = 0.0f;
    for (int j = 0; j < N; ++j) {
      float e = __expf(si * sh_o[j] - m);
      den += e;
      num += e * sh_s[j];
    }
    out[(size_t)b * N + i] = si + beta * num / den;
  }
}

// x6 = d3 @ W6.T + b6 ; normalize quaternion (first 4); write tuple (x7, x7)
__global__ void head_kernel(const float* __restrict__ d3, const float* __restrict__ W6,
                            const float* __restrict__ b6, float* __restrict__ out, int B) {
  __shared__ float sh[8];
  int b = blockIdx.x;
  int lane = threadIdx.x;
  if (lane < 7) {
    const float* row = d3 + (size_t)b * 64;
    const float* w   = W6 + (size_t)lane * 64;
    float acc = b6[lane];
    for (int k = 0; k < 64; ++k) acc += row[k] * w[k];
    sh[lane] = acc;
  }
  __syncthreads();
  if (lane < 7) {
    float nrm = sqrtf(sh[0]*sh[0] + sh[1]*sh[1] + sh[2]*sh[2] + sh[3]*sh[3]);
    float v = sh[lane];
    if (lane < 4) v = v / fmaxf(nrm, 1e-12f);
    out[(size_t)b * 7 + lane] = v;
    out[(size_t)B * 7 + (size_t)b * 7 + lane] = v;
  }
}

extern "C" void kernel_launch(void* const* d_in, const int* in_sizes, int n_in,
                              void* d_out, int out_size, void* d_ws, size_t ws_size,
                              hipStream_t stream) {
  const float* x    = (const float*)d_in[0];
  const float* y    = (const float*)d_in[1];
  const float* W1   = (const float*)d_in[2];
  const float* be1  = (const float*)d_in[3];
  const float* fcW1 = (const float*)d_in[4];
  const float* fcb1 = (const float*)d_in[5];
  const float* W2   = (const float*)d_in[6];
  const float* be2  = (const float*)d_in[7];
  const float* fcW2 = (const float*)d_in[8];
  const float* fcb2 = (const float*)d_in[9];
  const float* W3   = (const float*)d_in[10];
  const float* be3  = (const float*)d_in[11];
  const float* fcW3 = (const float*)d_in[12];
  const float* fcb3 = (const float*)d_in[13];
  const float* W4   = (const float*)d_in[14];
  const float* be4  = (const float*)d_in[15];
  const float* fcW4 = (const float*)d_in[16];
  const float* fcb4 = (const float*)d_in[17];
  const float* W6   = (const float*)d_in[18];
  const float* b6   = (const float*)d_in[19];

  float* ws = (float*)d_ws;
  float* m  = ws;                 // 512*1024 mish/concat staging
  float* qa = m  + 524288;        // 512*512 max, pre-attn qx
  float* qb = qa + 262144;        // pre-attn qy
  float* xa = qb + 262144;        // post-attn x
  float* ya = xa + 262144;        // post-attn y
  float* d0 = ya + 262144;        // 512*512
  float* d1 = d0 + 262144;        // 512*256
  float* d2 = d1 + 131072;        // 512*128
  float* d3 = d2 + 65536;         // 512*64

  const int B = 512;
  auto grid1d = [](int n) { return (n + 255) / 256; };
  auto gemm = [&](const float* A, const float* W, const float* bias, float* C, int N, int K) {
    int tiles = (B >> 5) * (N >> 5);           // 32x32 tiles per wave
    hipLaunchKernelGGL(gemm_wmma_f32, dim3((tiles + 3) / 4), dim3(128), 0, stream,
                       A, W, bias, C, B, N, K);
  };
  auto stage = [&](const float* xin, const float* yin, const float* W, const float* beta,
                   int Kin, int Nout) {
    hipLaunchKernelGGL(mish_kernel, dim3(grid1d(B * Kin)), dim3(256), 0, stream, xin, m, B * Kin);
    gemm(m, W, nullptr, qa, Nout, Kin);
    hipLaunchKernelGGL(mish_kernel, dim3(grid1d(B * Kin)), dim3(256), 0, stream, yin, m, B * Kin);
    gemm(m, W, nullptr, qb, Nout, Kin);
    hipLaunchKernelGGL(attn_kernel, dim3(B), dim3(256), 0, stream, qa, qb, beta, xa, Nout);
    hipLaunchKernelGGL(attn_kernel, dim3(B), dim3(256), 0, stream, qb, qa, beta, ya, Nout);
  };
  auto fc = [&](const float* dprev, const float* fcW, const float* fcb, float* dout,
                int N, int Nout) {
    hipLaunchKernelGGL(cat_mish_kernel, dim3(grid1d(B * 2 * N)), dim3(256), 0, stream,
                       xa, ya, dprev, m, N, B);
    gemm(m, fcW, fcb, dout, Nout, 2 * N);
  };

  stage(x,  y,  W1, be1, 1024, 512);
  fc(nullptr, fcW1, fcb1, d0, 512, 512);
  stage(xa, ya, W2, be2, 512, 256);
  fc(d0, fcW2, fcb2, d1, 256, 256);
  stage(xa, ya, W3, be3, 256, 128);
  fc(d1, fcW3, fcb3, d2, 128, 128);
  stage(xa, ya, W4, be4, 128, 64);
  fc(d2, fcW4, fcb4, d3, 64, 64);
  hipLaunchKernelGGL(head_kernel, dim3(B), dim3(32), 0, stream, d3, W6, b6, (float*)d_out, B);

  (void)in_sizes; (void)n_in; (void)out_size; (void)ws_size;
}